// MagnetLoss_11991548690919
// MI455X (gfx1250) — compile-verified
//
#include <hip/hip_runtime.h>
#include <math.h>

#define N_PTS 65536
#define DIM   512
#define KCL   256
#define ALPHA 1.0f
#define EPS   1e-8f
#define XSTR  516   // padded LDS row stride in floats (4-float pad -> conflict-free)

typedef float v2f __attribute__((ext_vector_type(2)));
typedef float v4f __attribute__((ext_vector_type(4)));
typedef float v8f __attribute__((ext_vector_type(8)));

// ---------------- Kernel A: per-cluster centers, ||c||^2, sigma_k ----------------
// One block per cluster (K=256 blocks x 256 threads). Deterministic (no atomics,
// fixed accumulation order). Rows are scanned in chunks of 256: one coalesced cid
// load + wave32 ballot -> 8 membership masks in LDS -> only matching rows touched.
__global__ void __launch_bounds__(256)
cluster_stats_kernel(const float* __restrict__ x,
                     const int*   __restrict__ y,
                     float* __restrict__ centers,
                     float* __restrict__ cnorm2,
                     float* __restrict__ sigma) {
    const int k   = blockIdx.x;
    const int tid = threadIdx.x;                 // 0..255, owns dims tid and tid+256
    const int wid = tid >> 5;

    __shared__ unsigned mask_lds[8];

    float a0 = 0.f, a1 = 0.f, sq = 0.f;
    int   nm = 0;

    for (int base = 0; base < N_PTS; base += 256) {
        const int cid = y[3 * (base + tid) + 2];
        const unsigned long long bal = __ballot(cid == k);
        if ((tid & 31) == 0) mask_lds[wid] = (unsigned)bal;
        __syncthreads();

        #pragma unroll
        for (int wv = 0; wv < 8; ++wv) {
            unsigned m = mask_lds[wv];           // uniform across block
            while (m) {                          // typically 0-2 iterations
                const int bit = __ffs(m) - 1;
                m &= m - 1;
                const int row = base + wv * 32 + bit;
                const float v0 = x[(size_t)row * DIM + tid];
                const float v1 = x[(size_t)row * DIM + tid + 256];
                a0 += v0; a1 += v1;
                sq += v0 * v0 + v1 * v1;
                ++nm;
            }
        }
        __syncthreads();                         // masks consumed before next chunk
    }

    const float fn = (float)nm;
    const float c0 = a0 / fn, c1 = a1 / fn;
    centers[(size_t)k * DIM + tid]       = c0;
    centers[(size_t)k * DIM + tid + 256] = c1;

    __shared__ float redA[256];
    __shared__ float redB[256];
    redA[tid] = c0 * c0 + c1 * c1;
    redB[tid] = sq;
    __syncthreads();
    for (int s = 128; s > 0; s >>= 1) {
        if (tid < s) { redA[tid] += redA[tid + s]; redB[tid] += redB[tid + s]; }
        __syncthreads();
    }
    if (tid == 0) {
        const float cn = redA[0];
        cnorm2[k] = cn;
        sigma[k]  = redB[0] - fn * cn;   // sum ||x - c||^2 over members
    }
}

// ---------------- Kernel B: S = sum sigma, scale = -1/(2 S^2) ----------------
__global__ void __launch_bounds__(256)
scale_kernel(const float* __restrict__ sigma, float* __restrict__ scale_out) {
    __shared__ float red[256];
    const int tid = threadIdx.x;
    red[tid] = sigma[tid];
    __syncthreads();
    for (int s = 128; s > 0; s >>= 1) {
        if (tid < s) red[tid] += red[tid + s];
        __syncthreads();
    }
    if (tid == 0) {
        const float S = red[0];
        scale_out[0] = -1.0f / (2.0f * S * S);
    }
}

// ---------------- Kernel C: fused GEMM (WMMA f32 16x16x4) + loss epilogue ----------------
// 64 threads = 2 waves per block; each wave computes 16 rows x all 256 clusters.
__global__ void __launch_bounds__(64)
magnet_main_kernel(const float* __restrict__ x,
                   const int*   __restrict__ y,
                   const int*   __restrict__ clab,
                   const float* __restrict__ centers,
                   const float* __restrict__ cnorm2,
                   const float* __restrict__ scale_p,
                   float* __restrict__ partials) {
    __shared__ float XT[2][16 * XSTR];
    __shared__ float xn_lds[2][16];
    __shared__ float wsum[2];

    const int tid = threadIdx.x;
    const int w   = tid >> 5;        // wave id in block
    const int l   = tid & 31;        // lane
    const int lg  = l >> 4;          // 16-lane group (0/1)
    const int ln  = l & 15;          // lane within group
    const int rowbase = blockIdx.x * 32 + w * 16;

    // ---- stage 16x512 X tile into LDS (coalesced float4) ----
    for (int m = 0; m < 16; ++m) {
        const float* src = x + (size_t)(rowbase + m) * DIM;
        #pragma unroll
        for (int c = 0; c < DIM; c += 128) {
            v4f v = *(const v4f*)(src + c + l * 4);
            *(v4f*)(&XT[w][m * XSTR + c + l * 4]) = v;
        }
    }
    __syncthreads();

    // ---- row norms ||x_m||^2 from LDS (2 lanes per row) ----
    {
        const int m = ln;
        float s = 0.f;
        for (int k = lg; k < DIM; k += 2) {
            const float v = XT[w][m * XSTR + k];
            s += v * v;
        }
        s += __shfl_xor(s, 16);
        if (l < 16) xn_lds[w][l] = s;
    }
    __syncthreads();

    // ---- per-row constants (C/D layout: VGPR r <-> row r + lg*8) ----
    const float scale = scale_p[0];
    float xnr[8];
    int   y0r[8], y2r[8];
    #pragma unroll
    for (int r = 0; r < 8; ++r) {
        const int row = r + lg * 8;
        xnr[r] = xn_lds[w][row];
        y0r[r] = y[3 * (size_t)(rowbase + row)];
        y2r[r] = y[3 * (size_t)(rowbase + row) + 2];
    }
    float minv[8], numa[8];
    #pragma unroll
    for (int r = 0; r < 8; ++r) { minv[r] = __builtin_inff(); numa[r] = 0.f; }

    // ---- 16 column tiles of 16 clusters each ----
    const int koff = lg * 2;   // A/B fragment K offset for this lane group
    for (int kt = 0; kt < 16; ++kt) {
        const int   col = kt * 16 + ln;          // cluster column this lane owns
        const float cn  = cnorm2[col];
        const int   cl0 = clab[3 * col];
        const int   cl2 = clab[3 * col + 2];
        const float* crow = centers + (size_t)col * DIM + koff;
        const float* arow = &XT[w][ln * XSTR + koff];

        v8f acc = {0.f, 0.f, 0.f, 0.f, 0.f, 0.f, 0.f, 0.f};
        #pragma unroll 8
        for (int k0 = 0; k0 < DIM; k0 += 4) {
            v2f a = *(const v2f*)(arow + k0);    // A: 16x4 f32 fragment (LDS)
            v2f b = *(const v2f*)(crow + k0);    // B: 4x16 f32 fragment (L2-resident)
            acc = __builtin_amdgcn_wmma_f32_16x16x4_f32(
                      false, a, false, b, (short)0, acc, false, false);
        }

        // fused epilogue for this 16x16 tile
        #pragma unroll
        for (int r = 0; r < 8; ++r) {
            float d2  = xnr[r] + cn - 2.0f * acc[r];
            d2        = fmaxf(d2, 0.0f);
            const float dis = scale * d2;
            numa[r] += (cl2 == y2r[r]) ? dis : 0.0f;
            minv[r]  = fminf(minv[r], (cl0 != y0r[r]) ? dis : __builtin_inff());
        }
    }

    // ---- reduce across the 16-lane group, compute per-row loss ----
    float mysum = 0.f;
    #pragma unroll
    for (int r = 0; r < 8; ++r) {
        float mn = minv[r];
        float nu = numa[r];
        #pragma unroll
        for (int m = 8; m >= 1; m >>= 1) {
            mn = fminf(mn, __shfl_xor(mn, m));
            nu += __shfl_xor(nu, m);
        }
        const float numer = expf(nu + ALPHA);
        const float denum = expf(mn);              // exp(+inf) -> inf -> ratio 0: matches ref
        float loss = -logf(numer / (denum + EPS) + EPS);
        loss = fmaxf(loss, 0.f);
        if (ln == 0) mysum += loss;
    }
    mysum += __shfl_xor(mysum, 16);
    if (l == 0) wsum[w] = mysum;
    __syncthreads();
    if (tid == 0) partials[blockIdx.x] = wsum[0] + wsum[1];
}

// ---------------- Kernel D: mean over partials ----------------
__global__ void __launch_bounds__(256)
finalize_kernel(const float* __restrict__ partials, int np, float* __restrict__ out) {
    __shared__ float red[256];
    const int tid = threadIdx.x;
    float s = 0.f;
    for (int i = tid; i < np; i += 256) s += partials[i];
    red[tid] = s;
    __syncthreads();
    for (int st = 128; st > 0; st >>= 1) {
        if (tid < st) red[tid] += red[tid + st];
        __syncthreads();
    }
    if (tid == 0) out[0] = red[0] / (float)N_PTS;
}

extern "C" void kernel_launch(void* const* d_in, const int* in_sizes, int n_in,
                              void* d_out, int out_size, void* d_ws, size_t ws_size,
                              hipStream_t stream) {
    const float* x    = (const float*)d_in[0];   // (N, D) fp32
    const int*   y    = (const int*)d_in[1];     // (N, 3) int32
    const int*   clab = (const int*)d_in[2];     // (K, 3) int32
    float* out = (float*)d_out;

    float* ws       = (float*)d_ws;
    float* centers  = ws;                              // K*D floats (512 KB)
    float* cnorm2   = centers + (size_t)KCL * DIM;     // K floats
    float* sigma    = cnorm2 + KCL;                    // K floats
    float* scale    = sigma + KCL;                     // 1 float (padded)
    float* partials = scale + 4;                       // N/32 floats

    (void)in_sizes; (void)n_in; (void)out_size; (void)ws_size;

    cluster_stats_kernel<<<KCL, 256, 0, stream>>>(x, y, centers, cnorm2, sigma);
    scale_kernel<<<1, 256, 0, stream>>>(sigma, scale);
    const int nblk = N_PTS / 32;
    magnet_main_kernel<<<nblk, 64, 0, stream>>>(x, y, clab, centers, cnorm2, scale, partials);
    finalize_kernel<<<1, 256, 0, stream>>>(partials, nblk, out);
}